// NetGCN_44186623541946
// MI455X (gfx1250) — compile-verified
//
#include <hip/hip_runtime.h>
#include <math.h>

typedef __attribute__((ext_vector_type(2))) float v2f;
typedef __attribute__((ext_vector_type(8))) float v8f;
typedef __attribute__((ext_vector_type(4))) float f4;

#define N_GRAPHS  64
#define DIM       128
#define OUTF      16

// ---------------------------------------------------------------------------
// GEMM: Y[nrows, NCOLS] = X[nrows, DIM] @ W[DIM, NCOLS], fp32 WMMA 16x16x4.
// NCOLS is a compile-time constant so every A/B load uses an immediate
// instruction offset (no per-iteration 64-bit address math).
// One wave per 16x16 output tile. nrows % 16 == 0 (100000 = 6250*16).
// A-frag (16x4 f32): lane<16 -> M=lane, K={k0,k0+1}; lane>=16 -> M=lane-16, K={k0+2,k0+3}
// B-frag (4x16 f32): symmetric (rows striped across lanes within a VGPR)
// D (16x16 f32, 8 VGPRs): vgpr r -> M = r + 8*(lane>>4), N = lane&15
// ---------------------------------------------------------------------------
template <int NCOLS>
__global__ __launch_bounds__(32) void gcn_gemm_f32(const float* __restrict__ X,
                                                   const float* __restrict__ W,
                                                   float* __restrict__ Y) {
  const int lane = threadIdx.x;
  const int half = lane >> 4;       // 0 or 1
  const int l15  = lane & 15;
  const int row0 = blockIdx.x * 16;
  const int col0 = blockIdx.y * 16;

  const float* __restrict__ Arow = X + (size_t)(row0 + l15) * DIM + 2 * half;
  const float* __restrict__ Bcol = W + (size_t)(2 * half) * NCOLS + col0 + l15;

  v8f acc = {};
#pragma unroll
  for (int k0 = 0; k0 < DIM; k0 += 4) {
    v2f a = *(const v2f*)(Arow + k0);            // global_load_b64, imm offset
    v2f b;
    b.x = Bcol[(size_t)k0 * NCOLS];              // imm offset (k0*NCOLS*4 < 2^24)
    b.y = Bcol[(size_t)(k0 + 1) * NCOLS];
    acc = __builtin_amdgcn_wmma_f32_16x16x4_f32(false, a, false, b,
                                                (short)0, acc, false, false);
  }

  float* __restrict__ Yt = Y + (size_t)(row0 + 8 * half) * NCOLS + col0 + l15;
#pragma unroll
  for (int r = 0; r < 8; ++r)
    Yt[(size_t)r * NCOLS] = acc[r];
}

// ---------------------------------------------------------------------------
// Edge aggregation: AGG[dst[e], :] += XW[src[e], :]. One wave per edge,
// each lane moves float4 (global_load_b128 gather, 4 atomic f32 adds/lane).
// Both 51MB buffers are L2-resident (192MB L2) -> atomics stay on-chip.
// ---------------------------------------------------------------------------
__global__ __launch_bounds__(256) void gcn_edge_scatter(const float* __restrict__ XW,
                                                        const int* __restrict__ src,
                                                        const int* __restrict__ dst,
                                                        float* __restrict__ AGG,
                                                        int nedges) {
  const int e = blockIdx.x * (blockDim.x >> 5) + (threadIdx.x >> 5);
  if (e >= nedges) return;
  const int lane = threadIdx.x & 31;
  const int s = __builtin_amdgcn_readfirstlane(src[e]);   // wave-uniform scalar
  const int d = __builtin_amdgcn_readfirstlane(dst[e]);
  f4 v = ((const f4*)(XW + (size_t)s * DIM))[lane];
  float* a = AGG + (size_t)d * DIM + lane * 4;
  atomicAdd(a + 0, v.x);
  atomicAdd(a + 1, v.y);
  atomicAdd(a + 2, v.z);
  atomicAdd(a + 3, v.w);
}

__global__ void gcn_fill0(float* __restrict__ p, long n) {
  long i = (long)blockIdx.x * blockDim.x + threadIdx.x;
  if (i < n) p[i] = 0.0f;
}

__global__ void gcn_relu(float* __restrict__ p, long n) {
  long i = (long)blockIdx.x * blockDim.x + threadIdx.x;
  if (i < n) p[i] = fmaxf(p[i], 0.0f);
}

// ---------------------------------------------------------------------------
// Per-graph sums + counts: one wave per node (float4 per lane).
// ---------------------------------------------------------------------------
__global__ __launch_bounds__(256) void gcn_pool(const float* __restrict__ H,
                                                const int* __restrict__ gid,
                                                float* __restrict__ PS,
                                                float* __restrict__ CNT,
                                                int nnodes) {
  const int node = blockIdx.x * (blockDim.x >> 5) + (threadIdx.x >> 5);
  if (node >= nnodes) return;
  const int lane = threadIdx.x & 31;
  const int g = __builtin_amdgcn_readfirstlane(gid[node]);
  f4 v = ((const f4*)(H + (size_t)node * DIM))[lane];
  float* p = PS + (size_t)g * DIM + lane * 4;
  atomicAdd(p + 0, v.x);
  atomicAdd(p + 1, v.y);
  atomicAdd(p + 2, v.z);
  atomicAdd(p + 3, v.w);
  if (lane == 0) atomicAdd(CNT + g, 1.0f);
}

// ---------------------------------------------------------------------------
// Final FC: out = sigmoid( (PS / max(cnt,1)) @ W3 ), 64x128 @ 128x16.
// 4 waves (one per 16-row tile), mean-division fused into A-frag load.
// ---------------------------------------------------------------------------
__global__ __launch_bounds__(32) void gcn_final(const float* __restrict__ PS,
                                                const float* __restrict__ CNT,
                                                const float* __restrict__ W3,
                                                float* __restrict__ OUTP) {
  const int lane = threadIdx.x;
  const int half = lane >> 4;
  const int l15  = lane & 15;
  const int row0 = blockIdx.x * 16;

  const float c   = fmaxf(CNT[row0 + l15], 1.0f);
  const float inv = 1.0f / c;
  const float* __restrict__ Arow = PS + (size_t)(row0 + l15) * DIM + 2 * half;
  const float* __restrict__ Bcol = W3 + (size_t)(2 * half) * OUTF + l15;

  v8f acc = {};
#pragma unroll
  for (int k0 = 0; k0 < DIM; k0 += 4) {
    v2f a = *(const v2f*)(Arow + k0);
    a.x *= inv;
    a.y *= inv;
    v2f b;
    b.x = Bcol[(size_t)k0 * OUTF];
    b.y = Bcol[(size_t)(k0 + 1) * OUTF];
    acc = __builtin_amdgcn_wmma_f32_16x16x4_f32(false, a, false, b,
                                                (short)0, acc, false, false);
  }

#pragma unroll
  for (int r = 0; r < 8; ++r) {
    const int row = row0 + r + 8 * half;
    const float x = acc[r];
    OUTP[row * OUTF + l15] = 1.0f / (1.0f + expf(-x));
  }
}

extern "C" void kernel_launch(void* const* d_in, const int* in_sizes, int n_in,
                              void* d_out, int out_size, void* d_ws, size_t ws_size,
                              hipStream_t stream) {
  const float* features = (const float*)d_in[0];   // [N, 128]
  const float* W1       = (const float*)d_in[1];   // [128, 128]
  const float* W2       = (const float*)d_in[2];   // [128, 128]
  const float* W3       = (const float*)d_in[3];   // [128, 16]
  const int*   src      = (const int*)d_in[4];     // [E]
  const int*   dst      = (const int*)d_in[5];     // [E]
  const int*   gid      = (const int*)d_in[6];     // [N]

  const int nedges = in_sizes[4];
  const int nnodes = in_sizes[6];
  const long nd = (long)nnodes * DIM;

  // workspace carve: two N*DIM ping-pong buffers + pool sums/counts
  float* bufA = (float*)d_ws;                       // GEMM outputs
  float* bufB = bufA + (size_t)nnodes * DIM;        // aggregation target
  float* PS   = bufB + (size_t)nnodes * DIM;        // [64, 128] graph sums
  float* CNT  = PS + (size_t)N_GRAPHS * DIM;        // [64]

  const dim3 gemmGrid(nnodes / 16, DIM / 16);
  const int edgeBlocks = (nedges + 7) / 8;          // 8 waves/block, wave per edge
  const int nodeBlocks = (nnodes + 7) / 8;
  const int fillBlocks = (int)((nd + 255) / 256);

  // conv1: XW1 = features @ W1 ; agg ; ReLU
  gcn_gemm_f32<DIM><<<gemmGrid, 32, 0, stream>>>(features, W1, bufA);
  gcn_fill0<<<fillBlocks, 256, 0, stream>>>(bufB, nd);
  gcn_edge_scatter<<<edgeBlocks, 256, 0, stream>>>(bufA, src, dst, bufB, nedges);
  gcn_relu<<<fillBlocks, 256, 0, stream>>>(bufB, nd);

  // conv2: XW2 = h1 @ W2 ; agg
  gcn_gemm_f32<DIM><<<gemmGrid, 32, 0, stream>>>(bufB, W2, bufA);
  gcn_fill0<<<fillBlocks, 256, 0, stream>>>(bufB, nd);
  gcn_edge_scatter<<<edgeBlocks, 256, 0, stream>>>(bufA, src, dst, bufB, nedges);

  // mean-pool + FC + sigmoid
  gcn_fill0<<<(N_GRAPHS * DIM + N_GRAPHS + 255) / 256, 256, 0, stream>>>(PS, (long)(N_GRAPHS * DIM + N_GRAPHS));
  gcn_pool<<<nodeBlocks, 256, 0, stream>>>(bufB, gid, PS, CNT, nnodes);
  gcn_final<<<N_GRAPHS / 16, 32, 0, stream>>>(PS, CNT, W3, (float*)d_out);
}